// HausdorffEditDistance_23021024707152
// MI455X (gfx1250) — compile-verified
//
#include <hip/hip_runtime.h>
#include <cstdint>

// ---------------------------------------------------------------------------
// Hausdorff Edit Distance, B=32, N=M=1024, D=2  — CDNA5 (gfx1250) WMMA path.
// dist2(n,m) = rsq[n] + csq[m] + (A x B)[n,m]   with A = -2*w*pos1, B = pos2.
// rsq+csq is folded into the WMMA C operand: D = A x B + C gives dist2 tiles
// directly via V_WMMA_F32_16X16X4_F32 (K padded 2->4 with zero lanes).
// ---------------------------------------------------------------------------

constexpr float kBeta = 0.1f;
constexpr float kTauN = 4.0f;
constexpr float kTauE = 16.0f;
constexpr float kBig  = 1e9f;
constexpr int   kB = 32, kN = 1024, kM = 1024;

typedef __attribute__((ext_vector_type(2))) float v2f;
typedef __attribute__((ext_vector_type(8))) float v8f;

// ---------------------------------------------------------------------------
// Kernel 1: seed row/col minima with deletion costs (bit-pattern ints).
// Padded rows/cols start at 0.0 and can never be lowered (all dists >= 0),
// which reproduces the reference's post-mask zeroing exactly.
// ---------------------------------------------------------------------------
__global__ void __launch_bounds__(256) hed_init_kernel(
    const int* __restrict__ deg1, const int* __restrict__ deg2,
    const int* __restrict__ n1, const int* __restrict__ n2,
    int* __restrict__ row_min, int* __restrict__ col_min)
{
  const int idx = blockIdx.x * 256 + threadIdx.x;   // 0 .. B*N-1   (N == M)
  const int b = idx >> 10;
  const int i = idx & 1023;
  const float d1c = kTauN + (float)deg1[b * kN + i] * (kTauE * 0.5f);
  const float d2c = kTauN + (float)deg2[b * kM + i] * (kTauE * 0.5f);
  row_min[idx] = __float_as_int(i < n1[b] ? d1c : 0.0f);
  col_min[idx] = __float_as_int(i < n2[b] ? d2c : 0.0f);
}

// ---------------------------------------------------------------------------
// Kernel 2: one wave owns a 16-row strip x 256 columns (16 WMMA tiles).
//   A operand (16x4 f32): lanes 0-15 hold K=0,1 = (-2*w0*x1, -2*w1*y1),
//                         lanes 16-31 hold K=2,3 = 0 (padding).
//   B operand (4x16 f32): vgpr0 lanes 0-15 = K0 row (x2),
//                         lanes 16-31 = K1 row (y2); vgpr1 (K2,K3) = 0.
//   C/D (16x16 f32): vgpr r, lanes 0-15 -> row r; lanes 16-31 -> row r+8.
// Masking is additive (+1e9) so the hot loop stays branch-free; padded
// rows/cols are pinned at 0 in the min buffers so their atomics are no-ops.
// ---------------------------------------------------------------------------
__global__ void __launch_bounds__(256) hed_tiles_kernel(
    const float* __restrict__ pos1, const float* __restrict__ pos2,
    const int* __restrict__ deg1, const int* __restrict__ deg2,
    const float* __restrict__ stdw, const int* __restrict__ n1,
    const int* __restrict__ n2, int* __restrict__ row_min,
    int* __restrict__ col_min)
{
  const int w    = blockIdx.x * 8 + (threadIdx.x >> 5);  // global wave id
  const int lane = threadIdx.x & 31;
  const int b         = w >> 8;          // 256 waves per batch
  const int rem       = w & 255;
  const int row_base  = (rem >> 2) << 4; // 64 row strips of 16
  const int col_start = (rem & 3) << 8;  // 4 column chunks of 256

  const int l  = lane & 15;
  const int hi = lane >> 4;              // 0 = lanes 0-15, 1 = lanes 16-31

  const float w0 = kBeta * stdw[b * 2 + 0];
  const float w1 = (1.0f - kBeta) * stdw[b * 2 + 1];

  const float2* p1 = reinterpret_cast<const float2*>(pos1) + b * kN;
  const float2* p2 = reinterpret_cast<const float2*>(pos2) + b * kM;

  const int nn1 = n1[b];
  const int nn2 = n2[b];

  // A operand, loaded once per wave (reused across all 16 column tiles).
  const float2 pa = p1[row_base + l];
  v2f amat;
  amat.x = hi ? 0.0f : (-2.0f * w0 * pa.x);
  amat.y = hi ? 0.0f : (-2.0f * w1 * pa.y);

  // Per-lane copies of the 8 row squared-norms / degrees / mask biases.
  float rsq[8], f1r[8], rbig[8];
  #pragma unroll
  for (int r = 0; r < 8; ++r) {
    const int row = row_base + r + hi * 8;
    const float2 q = p1[row];
    rsq[r]  = w0 * q.x * q.x + w1 * q.y * q.y;
    f1r[r]  = (float)deg1[b * kN + row];
    rbig[r] = (row < nn1) ? 0.0f : kBig;   // row-invalid bias (for col mins)
  }

  float rmin[8];
  #pragma unroll
  for (int r = 0; r < 8; ++r) rmin[r] = kBig;

  #pragma unroll 4
  for (int ct = 0; ct < 16; ++ct) {
    const int col = col_start + ct * 16 + l;
    const float2 pc = p2[col];

    v2f bmat;
    bmat.x = hi ? pc.y : pc.x;   // K=0 row = x2 (lanes 0-15), K=1 row = y2
    bmat.y = 0.0f;               // K=2,3 padding

    const float csq  = w0 * pc.x * pc.x + w1 * pc.y * pc.y;
    const float f2c  = (float)deg2[b * kM + col];
    const float cbig = (col < nn2) ? 0.0f : kBig;  // col-invalid bias

    // Seed the accumulator with rsq[row] + csq[col]; WMMA adds the cross term.
    v8f acc;
    #pragma unroll
    for (int r = 0; r < 8; ++r) acc[r] = rsq[r] + csq;

    acc = __builtin_amdgcn_wmma_f32_16x16x4_f32(
        /*neg_a=*/false, amat, /*neg_b=*/false, bmat,
        /*c_mod=*/(short)0, acc, /*reuse_a=*/false, /*reuse_b=*/false);

    float cmin = kBig;
    #pragma unroll
    for (int r = 0; r < 8; ++r) {
      const float d2 = fmaxf(acc[r], 0.0f);
      // (ALPHA*sqrt + TAU_E*|f1-f2|/2) / 2  ==  0.25*sqrt + 4*|f1-f2|
      const float v = 0.25f * __builtin_amdgcn_sqrtf(d2)
                    + 4.0f * __builtin_fabsf(f1r[r] - f2c);
      rmin[r] = fminf(rmin[r], v + cbig);     // col-masked for row mins
      cmin    = fminf(cmin,    v + rbig[r]);  // row-masked for col mins
    }
    // Column min over all 16 rows: fold the two half-waves together.
    cmin = fminf(cmin, __shfl_xor(cmin, 16, 32));
    if (hi == 0)
      atomicMin(&col_min[b * kM + col], __float_as_int(cmin));
  }

  // Row minima: reduce across the 16 lanes of each half, 1 atomic per row.
  #pragma unroll
  for (int r = 0; r < 8; ++r) {
    float v = rmin[r];
    v = fminf(v, __shfl_xor(v, 1, 32));
    v = fminf(v, __shfl_xor(v, 2, 32));
    v = fminf(v, __shfl_xor(v, 4, 32));
    v = fminf(v, __shfl_xor(v, 8, 32));
    if (l == r)
      atomicMin(&row_min[b * kN + row_base + r + hi * 8], __float_as_int(v));
  }
}

// ---------------------------------------------------------------------------
// Kernel 3: per-batch reduction -> final normalized distance.
// ---------------------------------------------------------------------------
__global__ void __launch_bounds__(256) hed_final_kernel(
    const int* __restrict__ row_min, const int* __restrict__ col_min,
    const int* __restrict__ deg1, const int* __restrict__ deg2,
    const int* __restrict__ n1, const int* __restrict__ n2,
    float* __restrict__ out)
{
  const int b   = blockIdx.x;
  const int tid = threadIdx.x;
  const int nn1 = n1[b], nn2 = n2[b];

  float s = 0.0f, e1 = 0.0f, e2 = 0.0f;
  for (int i = tid; i < kN; i += 256) {
    s += __int_as_float(row_min[b * kN + i]);
    if (i < nn1) e1 += (float)deg1[b * kN + i];
  }
  for (int i = tid; i < kM; i += 256) {
    s += __int_as_float(col_min[b * kM + i]);
    if (i < nn2) e2 += (float)deg2[b * kM + i];
  }

  __shared__ float red0[256], red1[256], red2[256];
  red0[tid] = s; red1[tid] = e1; red2[tid] = e2;
  __syncthreads();
  for (int off = 128; off > 0; off >>= 1) {
    if (tid < off) {
      red0[tid] += red0[tid + off];
      red1[tid] += red1[tid + off];
      red2[tid] += red2[tid + off];
    }
    __syncthreads();
  }

  if (tid == 0) {
    const float nf1 = (float)nn1, nf2 = (float)nn2;
    float d = fmaxf(red0[0], __builtin_fabsf(nf1 - nf2) * kTauN);
    const float norm = kTauN * (nf1 + nf2) + kTauE * (red1[0] + red2[0]);
    out[b] = d / norm;
  }
}

// ---------------------------------------------------------------------------
extern "C" void kernel_launch(void* const* d_in, const int* in_sizes, int n_in,
                              void* d_out, int out_size, void* d_ws, size_t ws_size,
                              hipStream_t stream) {
  (void)in_sizes; (void)n_in; (void)out_size; (void)ws_size;

  const float* pos1 = (const float*)d_in[0];
  const float* pos2 = (const float*)d_in[1];
  const int*   deg1 = (const int*)d_in[2];
  const int*   deg2 = (const int*)d_in[3];
  const float* stdw = (const float*)d_in[4];
  const int*   n1   = (const int*)d_in[5];
  const int*   n2   = (const int*)d_in[6];
  float*       out  = (float*)d_out;

  int* row_min = (int*)d_ws;            // B*N ints
  int* col_min = row_min + kB * kN;     // B*M ints

  // 1) seed minima with deletion costs / zeros for padded entries
  hipLaunchKernelGGL(hed_init_kernel, dim3((kB * kN) / 256), dim3(256), 0,
                     stream, deg1, deg2, n1, n2, row_min, col_min);

  // 2) WMMA distance tiles + atomic min reductions
  //    waves = B(32) * row-strips(64) * col-chunks(4) = 8192 -> 1024 blocks
  hipLaunchKernelGGL(hed_tiles_kernel, dim3(1024), dim3(256), 0, stream,
                     pos1, pos2, deg1, deg2, stdw, n1, n2, row_min, col_min);

  // 3) per-batch reduction + normalization
  hipLaunchKernelGGL(hed_final_kernel, dim3(kB), dim3(256), 0, stream,
                     row_min, col_min, deg1, deg2, n1, n2, out);
}